// EGTOCuda_86303072845865
// MI455X (gfx1250) — compile-verified
//
#include <hip/hip_runtime.h>
#include <math.h>

// ---------------- problem constants (match reference) ----------------
#define NATOM   32
#define NGAUSS  24
#define NORB    20          // 1 + 3 + 6 + 10 monomials for L = 0..3
#define NSPEC   4
#define NBODY   10          // 4 single + 6 cross
#define FP_SIZE 960         // NBODY * 4 * NGAUSS
#define PI_F    3.14159265358979323846f
#define ETA_F   1.2f
#define RCUT_F  6.0f

// padded LDS strides: even (keeps b64 8B alignment) and conflict-free
// (stride/2 coprime with 32 -> 16 distinct bank-pairs per half-wave)
#define PJ_STRIDE 34        // PmatT/RmatT fast dim (j = 0..31)
#define AC_STRIDE 82        // AmatT fast dim (c = 0..79)

typedef __attribute__((ext_vector_type(2))) float v2f;
typedef __attribute__((ext_vector_type(8))) float v8f;

// multinomial weights i!/(n!m!k!) per orbital, grouped L=0 | L=1 | L=2 | L=3
__constant__ float c_W[NORB] = {
    1.f,
    1.f, 1.f, 1.f,
    1.f, 2.f, 1.f, 2.f, 2.f, 1.f,
    1.f, 3.f, 3.f, 1.f, 3.f, 6.f, 3.f, 3.f, 3.f, 1.f};
__constant__ int c_LSTART[4] = {0, 1, 4, 10};
__constant__ int c_LEND[4]   = {1, 4, 10, 20};
// body -> species pair (first 4 are single-species, then COMBS order)
__constant__ int c_E1[NBODY] = {0, 1, 2, 3, 0, 0, 0, 1, 1, 2};
__constant__ int c_E2[NBODY] = {0, 1, 2, 3, 1, 2, 3, 2, 3, 3};

__global__ __launch_bounds__(32)
void egto_fp_kernel(const float* __restrict__ coords,   // [B][32][3]
                    const int*   __restrict__ Zc,       // [B][32]
                    const int*   __restrict__ counts,   // [B]
                    float*       __restrict__ out)      // [B][32][960]
{
    // monomial exponent tables (compile-time folded via full unroll)
    const int CN[NORB]  = {0, 1,0,0, 2,1,0,1,0,0, 3,2,1,0,2,1,0,1,0,0};
    const int CM[NORB]  = {0, 0,1,0, 0,1,2,0,1,0, 0,1,2,3,0,1,2,0,1,0};
    const int CK[NORB]  = {0, 0,0,1, 0,0,0,1,1,2, 0,0,0,0,1,1,1,2,2,3};
    const int OID[NORB] = {0, 1,1,1, 2,2,2,2,2,2, 3,3,3,3,3,3,3,3,3,3};

    // transposed staging: fast dim = j so each WMMA fragment is one ds_load_b64
    __shared__ float PmatT[NSPEC * NORB][PJ_STRIDE]; // [c][j] : ang * onehot
    __shared__ float RmatT[32][PJ_STRIDE];           // [g][j] : radial (g>=24 = 0)
    __shared__ float AmatT[32][AC_STRIDE];           // [g][c] : GEMM result

    const int lane = threadIdx.x;                // neighbor index j
    const int b    = blockIdx.x >> 5;
    const int i    = blockIdx.x & (NATOM - 1);
    const int cnt  = counts[b];

    // ---- per-pair geometry: lane j vs. center atom i ----
    const float xi = coords[((size_t)b * NATOM + i) * 3 + 0];
    const float yi = coords[((size_t)b * NATOM + i) * 3 + 1];
    const float zi = coords[((size_t)b * NATOM + i) * 3 + 2];
    const float xj = coords[((size_t)b * NATOM + lane) * 3 + 0];
    const float yj = coords[((size_t)b * NATOM + lane) * 3 + 1];
    const float zj = coords[((size_t)b * NATOM + lane) * 3 + 2];

    const float dx = xi - xj, dy = yi - yj, dz = zi - zj;
    const float d2 = dx * dx + dy * dy + dz * dz;
    const bool pm = (i < cnt) && (lane < cnt) && (lane != i) &&
                    (d2 < RCUT_F * RCUT_F);
    const float r    = sqrtf(pm ? d2 : 1.0f);
    const float cut  = pm ? 0.5f * (__cosf(r * (PI_F / RCUT_F)) + 1.0f) : 0.0f;
    const float rinv = 1.0f / r;

    // ---- radial column: R[g][j] = sqrt(eta/pi) * exp(-eta (r-off_g)^2) * cut ----
    const float rnorm = 0.61803865f; // sqrt(1.2/pi)
#pragma unroll
    for (int g = 0; g < NGAUSS; ++g) {
        const float t = r - 0.25f * (float)(g + 1);   // offsets 0.25 .. 6.0
        RmatT[g][lane] = rnorm * __expf(-ETA_F * t * t) * cut;
    }
#pragma unroll
    for (int g = NGAUSS; g < 32; ++g) RmatT[g][lane] = 0.0f;

    // ---- angular column: P[e*20+m][j] = x^n y^m z^k * r^-L in own species block ----
    const int zz = Zc[(size_t)b * NATOM + lane];
    const int e  = (zz == 1) ? 0 : (zz == 6) ? 1 : (zz == 7) ? 2 : (zz == 8) ? 3 : -1;

    float xp[4], yp[4], zp[4], rp[4];
    xp[0] = yp[0] = zp[0] = rp[0] = 1.0f;
#pragma unroll
    for (int t = 1; t < 4; ++t) {
        xp[t] = xp[t - 1] * dx;
        yp[t] = yp[t - 1] * dy;
        zp[t] = zp[t - 1] * dz;
        rp[t] = rp[t - 1] * rinv;
    }
    // zero own column, then fill own species block (same-lane LDS ops stay in order)
    for (int c = 0; c < NSPEC * NORB; ++c) PmatT[c][lane] = 0.0f;
    if (pm && e >= 0) {
#pragma unroll
        for (int m = 0; m < NORB; ++m)
            PmatT[e * NORB + m][lane] =
                xp[CN[m]] * yp[CM[m]] * zp[CK[m]] * rp[OID[m]];
    }
    __syncthreads();

    // ---- per-atom GEMM: A[80][32] = P^T (80x32) x R (32x32) via WMMA f32 16x16x4 ----
    // A-frag 16x4: lanes 0-15 carry K=0/1 (v0/v1), lanes 16-31 carry K=2/3, M = lane&15
    // B-frag 4x16: VGPR v = row K=v (lanes 0-15) / K=v+2 (lanes 16-31), N = lane&15
    // D-tile 16x16: VGPR r = row M=r (lanes 0-15) / M=r+8 (lanes 16-31), N = lane&15
    const int hs  = lane >> 4;     // half-wave select
    const int l16 = lane & 15;
#pragma unroll
    for (int mt = 0; mt < 5; ++mt) {                 // 5 tiles of 16 over M=80
        v8f acc0 = {0.f, 0.f, 0.f, 0.f, 0.f, 0.f, 0.f, 0.f};
        v8f acc1 = {0.f, 0.f, 0.f, 0.f, 0.f, 0.f, 0.f, 0.f};
#pragma unroll
        for (int k = 0; k < 32; k += 4) {            // 8 K-steps over neighbors
            const int ka = k + (hs << 1);            // even -> 8B-aligned b64 loads
            const v2f af = *(const v2f*)&PmatT[mt * 16 + l16][ka]; // A[M,K..K+1]
            const v2f b0 = *(const v2f*)&RmatT[l16][ka];           // B[K..K+1, N]
            const v2f b1 = *(const v2f*)&RmatT[16 + l16][ka];      // N-tile 1
            acc0 = __builtin_amdgcn_wmma_f32_16x16x4_f32(
                false, af, false, b0, (short)0, acc0, false, false);
            acc1 = __builtin_amdgcn_wmma_f32_16x16x4_f32(
                false, af, false, b1, (short)0, acc1, false, false);
        }
#pragma unroll
        for (int rr = 0; rr < 8; rr += 2) {          // paired b64 stores of D
            const int crow = mt * 16 + (hs << 3) + rr;
            v2f d0 = {acc0[rr], acc0[rr + 1]};
            v2f d1 = {acc1[rr], acc1[rr + 1]};
            *(v2f*)&AmatT[l16][crow]      = d0;      // g = l16
            *(v2f*)&AmatT[16 + l16][crow] = d1;      // g = 16 + l16
        }
    }
    __syncthreads();

    // ---- epilogue: fp[body][l][g] = (2 if cross)*sum_m w[m] A[e1,m,g] A[e2,m,g] ----
    float* op = out + (size_t)blockIdx.x * FP_SIZE;
    for (int o = lane; o < FP_SIZE; o += 32) {
        const int body = o / (4 * NGAUSS);
        const int rem  = o - body * (4 * NGAUSS);
        const int l    = rem / NGAUSS;
        const int g    = rem - l * NGAUSS;
        const int e1   = c_E1[body] * NORB;
        const int e2   = c_E2[body] * NORB;
        float s = 0.0f;
        for (int m = c_LSTART[l]; m < c_LEND[l]; ++m)
            s += c_W[m] * AmatT[g][e1 + m] * AmatT[g][e2 + m];
        op[o] = (body < NSPEC) ? s : 2.0f * s;
    }
}

extern "C" void kernel_launch(void* const* d_in, const int* in_sizes, int n_in,
                              void* d_out, int out_size, void* d_ws, size_t ws_size,
                              hipStream_t stream) {
    (void)n_in; (void)out_size; (void)d_ws; (void)ws_size;
    const float* coords = (const float*)d_in[0];
    const int*   Zc     = (const int*)d_in[1];
    const int*   counts = (const int*)d_in[2];
    float*       out    = (float*)d_out;
    const int B = in_sizes[2];                 // natom_counts has B elements
    egto_fp_kernel<<<dim3(B * NATOM), dim3(32), 0, stream>>>(coords, Zc, counts, out);
}